// DynamicSubgraphGCN_5102421147741
// MI455X (gfx1250) — compile-verified
//
#include <hip/hip_runtime.h>
#include <math.h>

#define B_   16
#define N_   2048
#define D_   64
#define E_   32768
#define EPS_ 1e-6f

typedef float v2f __attribute__((ext_vector_type(2)));
typedef float v8f __attribute__((ext_vector_type(8)));

// ---------------- degree / normalization ----------------

__global__ void k_init_deg(float* __restrict__ deg) {
    int i = blockIdx.x * blockDim.x + threadIdx.x;
    if (i < B_ * N_) deg[i] = 1.0f;   // self-loop from A + I
}

__global__ void k_edge_deg(const int* __restrict__ ei, const float* __restrict__ m,
                           float* __restrict__ deg) {
    int t = blockIdx.x * blockDim.x + threadIdx.x;
    if (t >= B_ * E_) return;
    int b = t / E_, e = t - b * E_;
    int s = ei[b * 2 * E_ + e];
    int d = ei[b * 2 * E_ + E_ + e];
    float w = m[b * E_ + e];
    atomicAdd(&deg[b * N_ + s], w);
    atomicAdd(&deg[b * N_ + d], w);
}

__global__ void k_finalize_dis(float* __restrict__ dis) {
    int i = blockIdx.x * blockDim.x + threadIdx.x;
    if (i < B_ * N_) dis[i] = rsqrtf(fmaxf(dis[i], EPS_));
}

// ---------------- sparse aggregation: Y = A_norm * X ----------------

// Self-loop term initializes Y:  Y[b,n,:] = dis[b,n]^2 * X[b,n,:]
__global__ void k_agg_self(const float* __restrict__ X, const float* __restrict__ dis,
                           float* __restrict__ Y) {
    int i = blockIdx.x * blockDim.x + threadIdx.x;   // float4 granularity
    if (i >= B_ * N_ * D_ / 4) return;
    int node = i / (D_ / 4);                         // b*N_ + n
    float s = dis[node];
    s = s * s;
    float4 x = ((const float4*)X)[i];
    float4 y;
    y.x = s * x.x; y.y = s * x.y; y.z = s * x.z; y.w = s * x.w;
    ((float4*)Y)[i] = y;
}

// Edge scatter: 16 threads per edge, 4 consecutive d's per thread.
__global__ void k_agg_edges(const float* __restrict__ X, const int* __restrict__ ei,
                            const float* __restrict__ m, const float* __restrict__ dis,
                            float* __restrict__ Y) {
    long long t = (long long)blockIdx.x * blockDim.x + threadIdx.x;
    if (t >= (long long)B_ * E_ * 16) return;
    int chunk = (int)(t & 15);
    int eg    = (int)(t >> 4);
    int b = eg / E_, e = eg - b * E_;
    int s = ei[b * 2 * E_ + e];
    int d = ei[b * 2 * E_ + E_ + e];
    float w = m[b * E_ + e] * dis[b * N_ + s] * dis[b * N_ + d];

    const float4* Xs = (const float4*)(X + (size_t)(b * N_ + s) * D_);
    const float4* Xd = (const float4*)(X + (size_t)(b * N_ + d) * D_);
    float4 xs = Xs[chunk];
    float4 xd = Xd[chunk];
    float* Ys = Y + (size_t)(b * N_ + s) * D_ + chunk * 4;
    float* Yd = Y + (size_t)(b * N_ + d) * D_ + chunk * 4;
    atomicAdd(Yd + 0, w * xs.x); atomicAdd(Yd + 1, w * xs.y);
    atomicAdd(Yd + 2, w * xs.z); atomicAdd(Yd + 3, w * xs.w);
    atomicAdd(Ys + 0, w * xd.x); atomicAdd(Ys + 1, w * xd.y);
    atomicAdd(Ys + 2, w * xd.z); atomicAdd(Ys + 3, w * xd.w);
}

// ---------------- WMMA feature GEMM:  Out[n,e] = sum_d Y[n,d] * W[e,d]  (+GELU) ----

__global__ __launch_bounds__(256) void k_gemm_wmma(const float* __restrict__ Y,
                                                   const float* __restrict__ W,
                                                   float* __restrict__ Out,
                                                   int do_gelu) {
    __shared__ float sW[D_ * D_];   // 16 KB: full 64x64 weight tile
    int lin = threadIdx.y * 32 + threadIdx.x;
    {
        const float4* Wv  = (const float4*)W;
        float4*       sWv = (float4*)sW;
#pragma unroll
        for (int i = 0; i < 4; ++i) sWv[lin + 256 * i] = Wv[lin + 256 * i];
    }
    __syncthreads();

    int t     = blockIdx.x * 8 + threadIdx.y;  // global wave id: 16 batches * 128 row tiles
    int b     = t >> 7;
    int ntile = t & 127;
    int lane  = threadIdx.x;
    int half  = lane >> 4;    // 0: K pair (k0,k0+1) / M rows 0-7 of D; 1: (k0+2,k0+3) / rows 8-15
    int lrow  = lane & 15;

    const float* Yrow = Y + ((size_t)(b * N_) + (size_t)ntile * 16 + lrow) * D_;

    v8f acc[4] = {};  // 4 tiles of 16 output columns each -> full 64-wide row block

#pragma unroll
    for (int k0 = 0; k0 < D_; k0 += 4) {
        int kk = k0 + 2 * half;
        v2f a = *(const v2f*)(Yrow + kk);               // A: 16x4 f32 fragment
#pragma unroll
        for (int j = 0; j < 4; ++j) {
            // B[k][n] = W[e = j*16+lrow, d = kk(+1)]  -> contiguous float2 in LDS
            v2f bf = *(const v2f*)(sW + (size_t)(j * 16 + lrow) * D_ + kk);
            acc[j] = __builtin_amdgcn_wmma_f32_16x16x4_f32(
                false, a, false, bf, (short)0, acc[j], false, false);
        }
    }

#pragma unroll
    for (int j = 0; j < 4; ++j) {
#pragma unroll
        for (int r = 0; r < 8; ++r) {
            float v = acc[j][r];
            if (do_gelu) v = 0.5f * v * (1.0f + erff(v * 0.70710678118654752f));
            Out[((size_t)(b * N_) + (size_t)ntile * 16 + r + 8 * half) * D_ + j * 16 + lrow] = v;
        }
    }
}

// ---------------- launch ----------------

extern "C" void kernel_launch(void* const* d_in, const int* in_sizes, int n_in,
                              void* d_out, int out_size, void* d_ws, size_t ws_size,
                              hipStream_t stream) {
    const float* H  = (const float*)d_in[0];   // (B,N,D)
    const int*   ei = (const int*)d_in[1];     // (B,2,E)
    const float* m  = (const float*)d_in[2];   // (B,E)
    const float* W  = (const float*)d_in[3];   // (L,D,D)
    float* out = (float*)d_out;                // (B,N,D)

    float* ws  = (float*)d_ws;
    float* dis = ws;                           // B*N        = 32768 floats
    float* Y   = ws + 32768;                   // B*N*D      = 2097152 floats
    float* Xt  = Y + (size_t)B_ * N_ * D_;     // B*N*D      = 2097152 floats

    const int T = 256;

    // normalization vector
    k_init_deg<<<(B_ * N_ + T - 1) / T, T, 0, stream>>>(dis);
    k_edge_deg<<<(B_ * E_ + T - 1) / T, T, 0, stream>>>(ei, m, dis);
    k_finalize_dis<<<(B_ * N_ + T - 1) / T, T, 0, stream>>>(dis);

    dim3 gblk(32, 8);

    // layer 0: Y = A_norm * H ; Xt = gelu(Y * W0^T)
    k_agg_self<<<(B_ * N_ * D_ / 4 + T - 1) / T, T, 0, stream>>>(H, dis, Y);
    k_agg_edges<<<(B_ * E_ * 16 + T - 1) / T, T, 0, stream>>>(H, ei, m, dis, Y);
    k_gemm_wmma<<<256, gblk, 0, stream>>>(Y, W, Xt, 1);

    // layer 1: Y = A_norm * Xt ; out = Y * W1^T
    k_agg_self<<<(B_ * N_ * D_ / 4 + T - 1) / T, T, 0, stream>>>(Xt, dis, Y);
    k_agg_edges<<<(B_ * E_ * 16 + T - 1) / T, T, 0, stream>>>(Xt, ei, m, dis, Y);
    k_gemm_wmma<<<256, gblk, 0, stream>>>(Y, W + D_ * D_, out, 0);
}